// PointSample_36541581754600
// MI455X (gfx1250) — compile-verified
//
#include <hip/hip_runtime.h>

// Bilinear point sampling for MI455X (gfx1250, wave32).
// features: [8,128,128,256] f32, grid: [8,8192,2] f32 -> out: [8,8192,256] f32
//
// Roofline: ~134 MFLOP vs >=131 MB of traffic -> purely memory-bound
// (~5.6us floor @ 23.3 TB/s). No legal WMMA mapping exists (the 4-corner
// blend's "B operand" is gathered per point), so the optimal MI455X path is
// minimal-count, maximal-width coalesced VMEM:
//   - one point per wave32: all addresses/weights wave-uniform, zero divergence
//   - 8x global_load_b128 (512B/instr) cover the 4x1KB corner vectors exactly
//   - 2x global_store_b128 with TH=NT: output is write-once, keep it out of
//     the 192MB L2 so the 128MB features tensor stays resident for the gathers
//   - grid loads TH=NT (read-once stream)

// native clang vector type: valid for __builtin_nontemporal_{load,store}
typedef float vfloat4 __attribute__((ext_vector_type(4)));

namespace {
constexpr int B = 8;
constexpr int H = 128;
constexpr int W = 128;
constexpr int C = 256;
constexpr int P = 8192;
constexpr int WAVES_PER_BLOCK = 8;           // 256 threads = 8 wave32
constexpr int THREADS = WAVES_PER_BLOCK * 32;
} // namespace

__device__ __forceinline__ vfloat4 wsum4(vfloat4 a, vfloat4 b, vfloat4 c, vfloat4 d,
                                         float w0, float w1, float w2, float w3) {
  vfloat4 o;
  o.x = fmaf(d.x, w3, fmaf(c.x, w2, fmaf(b.x, w1, a.x * w0)));
  o.y = fmaf(d.y, w3, fmaf(c.y, w2, fmaf(b.y, w1, a.y * w0)));
  o.z = fmaf(d.z, w3, fmaf(c.z, w2, fmaf(b.z, w1, a.z * w0)));
  o.w = fmaf(d.w, w3, fmaf(c.w, w2, fmaf(b.w, w1, a.w * w0)));
  return o;
}

__global__ __launch_bounds__(THREADS) void point_sample_bilinear(
    const float* __restrict__ feat,
    const float* __restrict__ grid,
    float* __restrict__ out) {
  const int lane = threadIdx.x & 31;
  const int wave = threadIdx.x >> 5;
  const int pt = blockIdx.x * WAVES_PER_BLOCK + wave;  // global point id
  if (pt >= B * P) return;
  const int b = pt >> 13;  // / 8192

  // grid holds (x-like, y-like); reference reverses -> row = g.y*H-0.5, col = g.x*W-0.5
  // read-once stream: non-temporal load
  const float gxv = __builtin_nontemporal_load(grid + (size_t)pt * 2 + 0);
  const float gyv = __builtin_nontemporal_load(grid + (size_t)pt * 2 + 1);
  const float gr = fmaf(gyv, (float)H, -0.5f);
  const float gc = fmaf(gxv, (float)W, -0.5f);
  const float fr0 = floorf(gr);
  const float fc0 = floorf(gc);
  const float fy = gr - fr0;
  const float fx = gc - fc0;
  const int r0 = (int)fr0, c0 = (int)fc0;
  const int r1 = r0 + 1, c1 = c0 + 1;

  const bool r0ok = (unsigned)r0 < (unsigned)H;
  const bool r1ok = (unsigned)r1 < (unsigned)H;
  const bool c0ok = (unsigned)c0 < (unsigned)W;
  const bool c1ok = (unsigned)c1 < (unsigned)W;

  // corner weights, matching reference naming:
  // (r0,c0):(1-fy)(1-fx)  (r1,c0):fy(1-fx)  (r0,c1):(1-fy)fx  (r1,c1):fy*fx
  float w00 = (1.0f - fy) * (1.0f - fx);
  float w10 = fy * (1.0f - fx);
  float w01 = (1.0f - fy) * fx;
  float w11 = fy * fx;
  // zero-pad semantics: out-of-bounds corner contributes 0 (clamp addr, kill weight)
  w00 = (r0ok && c0ok) ? w00 : 0.0f;
  w10 = (r1ok && c0ok) ? w10 : 0.0f;
  w01 = (r0ok && c1ok) ? w01 : 0.0f;
  w11 = (r1ok && c1ok) ? w11 : 0.0f;

  const int rc0 = min(max(r0, 0), H - 1);
  const int rc1 = min(max(r1, 0), H - 1);
  const int cc0 = min(max(c0, 0), W - 1);
  const int cc1 = min(max(c1, 0), W - 1);

  const float* fb = feat + (size_t)b * (H * W * C);
  const float* p00 = fb + ((size_t)rc0 * W + cc0) * C;
  const float* p10 = fb + ((size_t)rc1 * W + cc0) * C;
  const float* p01 = fb + ((size_t)rc0 * W + cc1) * C;
  const float* p11 = fb + ((size_t)rc1 * W + cc1) * C;

  // lane covers channels [lane*4, lane*4+4) and [lane*4+128, lane*4+132):
  // each vfloat4 load is a contiguous 512B wavefront transaction (default RT
  // caching: the 128MB features tensor is the reuse stream -> keep in L2).
  const int off0 = lane * 4;
  const int off1 = off0 + 128;

  const vfloat4 a0 = *reinterpret_cast<const vfloat4*>(p00 + off0);
  const vfloat4 b0 = *reinterpret_cast<const vfloat4*>(p10 + off0);
  const vfloat4 e0 = *reinterpret_cast<const vfloat4*>(p01 + off0);
  const vfloat4 d0 = *reinterpret_cast<const vfloat4*>(p11 + off0);
  const vfloat4 a1 = *reinterpret_cast<const vfloat4*>(p00 + off1);
  const vfloat4 b1 = *reinterpret_cast<const vfloat4*>(p10 + off1);
  const vfloat4 e1 = *reinterpret_cast<const vfloat4*>(p01 + off1);
  const vfloat4 d1 = *reinterpret_cast<const vfloat4*>(p11 + off1);

  const vfloat4 o0 = wsum4(a0, b0, e0, d0, w00, w10, w01, w11);
  const vfloat4 o1 = wsum4(a1, b1, e1, d1, w00, w10, w01, w11);

  // write-once output: non-temporal b128 stores (TH=NT) so the 67MB output
  // stream doesn't evict the features working set from L2.
  float* ob = out + (size_t)pt * C;
  __builtin_nontemporal_store(o0, reinterpret_cast<vfloat4*>(ob + off0));
  __builtin_nontemporal_store(o1, reinterpret_cast<vfloat4*>(ob + off1));
}

extern "C" void kernel_launch(void* const* d_in, const int* in_sizes, int n_in,
                              void* d_out, int out_size, void* d_ws, size_t ws_size,
                              hipStream_t stream) {
  const float* feat = (const float*)d_in[0];  // [8,128,128,256] f32
  const float* grid = (const float*)d_in[1];  // [8,8192,2] f32
  float* out = (float*)d_out;                 // [8,8192,256] f32

  const int total_pts = B * P;                       // 65536 points
  const int blocks = total_pts / WAVES_PER_BLOCK;    // 8192 blocks, 1 pt/wave
  point_sample_bilinear<<<blocks, THREADS, 0, stream>>>(feat, grid, out);
}